// HierarchicalSoftmax_82454782148965
// MI455X (gfx1250) — compile-verified
//
#include <hip/hip_runtime.h>

typedef __attribute__((ext_vector_type(2))) float v2f;
typedef __attribute__((ext_vector_type(8))) float v8f;

#define VOCAB   100000
#define EMBED   128
#define BATCH   16384
#define LPATH   17
#define EPSF    1e-9f

#define TILE            16                          // batch rows per wave (WMMA M/N)
#define WAVES_PER_BLOCK 8
#define NTILES          (BATCH / TILE)              // 1024
#define TOTAL_WAVES     (NTILES * LPATH)            // 17408: one wave per (tile, level)
#define NBLOCKS         (TOTAL_WAVES / WAVES_PER_BLOCK) // 2176
#define LDS_STRIDE      9                           // pad 8 -> 9 floats/lane: conflict-free
#define REDUCE_THREADS  256

__global__ __launch_bounds__(WAVES_PER_BLOCK * 32)
void hs_loss_wmma_kernel(const int*   __restrict__ center_idx,
                         const int*   __restrict__ path_indices,
                         const int*   __restrict__ code_bits,
                         const float* __restrict__ center_emb,
                         const float* __restrict__ node_emb,
                         float*       __restrict__ block_partials)
{
    __shared__ float diag_lds[WAVES_PER_BLOCK][32 * LDS_STRIDE];
    __shared__ float wsum[WAVES_PER_BLOCK];

    const int lane = threadIdx.x & 31;
    const int wave = threadIdx.x >> 5;
    const int gw   = blockIdx.x * WAVES_PER_BLOCK + wave;  // global wave id
    const int tile = gw / LPATH;                           // batch tile
    const int l    = gw - tile * LPATH;                    // tree level
    const int b0   = tile * TILE;

    const int row = lane & 15;   // matrix row this lane loads fragments for
    const int hi  = lane >> 4;   // 0: K sub-cols {0,1}; 1: K sub-cols {2,3}

    // ---- Gather indices / bit for this (tile, level) ----------------------
    const int   cidx = center_idx[b0 + row];
    const int   nidx = path_indices[(b0 + row) * LPATH + l];
    const float bitv = (float)code_bits[(b0 + row) * LPATH + l];

    const float* crow = center_emb + (size_t)cidx * EMBED;
    const float* nrow = node_emb   + (size_t)nidx * EMBED;

    // ---- A fragments (center rows): issue all 32 loads up front -----------
    // A[16x4] chunk kc: lane L holds C[row][4*kc + 2*hi + {0,1}]
    v2f a[32];
#pragma unroll
    for (int kc = 0; kc < 32; ++kc)
        a[kc] = *(const v2f*)(crow + 4 * kc + 2 * hi);

    // ---- Single straight-line 32-step WMMA chain (K = 128) ----------------
    // B[4x16] chunk kc: lane L holds N[row][4*kc + 2*hi + {0,1}]
    v8f acc = {};
#pragma unroll
    for (int kc = 0; kc < 32; ++kc) {
        v2f bfrag = *(const v2f*)(nrow + 4 * kc + 2 * hi);
        // D(16x16,f32) += A(16x4,f32) * B(4x16,f32); diagonal = the 16 dots
        acc = __builtin_amdgcn_wmma_f32_16x16x4_f32(
                  false, a[kc], false, bfrag, (short)0, acc, false, false);
    }

    // ---- Extract diagonal via padded LDS ----------------------------------
    float* ld = &diag_lds[wave][0];
#pragma unroll
    for (int r = 0; r < 8; ++r)
        ld[lane * LDS_STRIDE + r] = acc[r];
    __builtin_amdgcn_wave_barrier();   // LDS ops are in-order within a wave

    float lane_loss = 0.0f;
    if (lane < 16) {
        // diag i<8 : VGPR i,   lane i ; diag i>=8 : VGPR i-8, lane i+16
        const int src_lane = (lane < 8) ? lane : (lane + 16);
        const int r        = lane & 7;
        const float dot    = ld[src_lane * LDS_STRIDE + r];

        const float s = 1.0f / (1.0f + __expf(-dot));
        lane_loss = -(bitv * __logf(s + EPSF) +
                      (1.0f - bitv) * __logf(1.0f - s + EPSF));
    }

    // ---- Wave reduction (lanes >=16 hold 0) -------------------------------
#pragma unroll
    for (int off = 16; off >= 1; off >>= 1)
        lane_loss += __shfl_xor(lane_loss, off, 32);
    if (lane == 0) wsum[wave] = lane_loss;
    __syncthreads();

    // ---- Block reduction, deterministic, no atomics -----------------------
    if (threadIdx.x == 0) {
        float t = 0.0f;
#pragma unroll
        for (int w = 0; w < WAVES_PER_BLOCK; ++w) t += wsum[w];
        block_partials[blockIdx.x] = t;
    }
}

__global__ __launch_bounds__(REDUCE_THREADS)
void hs_reduce_kernel(const float* __restrict__ partials,
                      float*       __restrict__ out, int n)
{
    __shared__ float s[REDUCE_THREADS];
    const int t = threadIdx.x;
    float v = 0.0f;
    for (int i = t; i < n; i += REDUCE_THREADS)   // fixed order per thread: deterministic
        v += partials[i];
    s[t] = v;
    __syncthreads();
#pragma unroll
    for (int off = REDUCE_THREADS / 2; off >= 1; off >>= 1) {
        if (t < off) s[t] += s[t + off];
        __syncthreads();
    }
    if (t == 0) out[0] = s[0];
}

extern "C" void kernel_launch(void* const* d_in, const int* in_sizes, int n_in,
                              void* d_out, int out_size, void* d_ws, size_t ws_size,
                              hipStream_t stream)
{
    const int*   center_idx   = (const int*)  d_in[0];
    const int*   path_indices = (const int*)  d_in[1];
    const int*   code_bits    = (const int*)  d_in[2];
    const float* center_emb   = (const float*)d_in[3];
    const float* node_emb     = (const float*)d_in[4];
    float*       out          = (float*)d_out;
    float*       partials     = (float*)d_ws;   // NBLOCKS floats

    hs_loss_wmma_kernel<<<NBLOCKS, WAVES_PER_BLOCK * 32, 0, stream>>>(
        center_idx, path_indices, code_bits, center_emb, node_emb, partials);
    hs_reduce_kernel<<<1, REDUCE_THREADS, 0, stream>>>(partials, out, NBLOCKS);
}